// SimpleNet_62105227100335
// MI455X (gfx1250) — compile-verified
//
#include <hip/hip_runtime.h>

typedef __attribute__((ext_vector_type(2))) float v2f;
typedef __attribute__((ext_vector_type(8))) float v8f;

// ---------------------------------------------------------------------------
// Scatter-mean accumulation over edges, both directions at once.
// One wave (32 lanes) per edge; float4 coalesced gathers, f32 global atomics.
// ---------------------------------------------------------------------------
__global__ __launch_bounds__(256) void scatter_edges(
    const float* __restrict__ xsU,  // features of user-side nodes [NU x D]
    const float* __restrict__ xsI,  // features of item-side nodes [NI x D]
    const int* __restrict__ src, const int* __restrict__ dst,
    float* __restrict__ aggI, float* __restrict__ aggU,
    float* __restrict__ degI, float* __restrict__ degU,
    int E, int D, int countDeg)
{
    int wid  = (blockIdx.x * blockDim.x + threadIdx.x) >> 5;
    int lane = threadIdx.x & 31;
    if (wid >= E) return;
    int s = src[wid];
    int d = dst[wid];
    if (countDeg && lane == 0) {
        atomicAdd(&degI[d], 1.0f);
        atomicAdd(&degU[s], 1.0f);
    }
    const float* xu = xsU + (size_t)s * D;
    const float* xi = xsI + (size_t)d * D;
    float* ai = aggI + (size_t)d * D;
    float* au = aggU + (size_t)s * D;
    for (int j = lane * 4; j < D; j += 32 * 4) {
        float4 vu = *(const float4*)(xu + j);
        atomicAdd(ai + j + 0, vu.x);
        atomicAdd(ai + j + 1, vu.y);
        atomicAdd(ai + j + 2, vu.z);
        atomicAdd(ai + j + 3, vu.w);
        float4 vi = *(const float4*)(xi + j);
        atomicAdd(au + j + 0, vi.x);
        atomicAdd(au + j + 1, vi.y);
        atomicAdd(au + j + 2, vi.z);
        atomicAdd(au + j + 3, vi.w);
    }
}

// ---------------------------------------------------------------------------
// agg[n][:] /= max(deg[n],1)
// ---------------------------------------------------------------------------
__global__ __launch_bounds__(256) void normalize_kernel(
    float* __restrict__ agg, const float* __restrict__ deg, int n, int D)
{
    int i = blockIdx.x * blockDim.x + threadIdx.x;
    int d4 = D >> 2;
    int total = n * d4;
    if (i >= total) return;
    int node = i / d4;
    float inv = 1.0f / fmaxf(deg[node], 1.0f);
    float4* p = (float4*)agg + i;
    float4 v = *p;
    v.x *= inv; v.y *= inv; v.z *= inv; v.w *= inv;
    *p = v;
}

// ---------------------------------------------------------------------------
// C[M,N] = act(A1[M,K] @ W1[K,N] + A2[M,K] @ W2[K,N] + bias[N])
// Block: 256 threads = 8 waves; block tile 64x128; wave tile 16x64.
// Exact fp32 matrix math via v_wmma_f32_16x16x4_f32.
// Both A and W tiles staged in LDS; inner loop is pure DS + WMMA.
// Requires N % 128 == 0 and K % 16 == 0 (true here: N in {128,256}, K in {128,256}).
// ---------------------------------------------------------------------------
__global__ __launch_bounds__(256) void gemm_dual(
    const float* __restrict__ A1, const float* __restrict__ W1,
    const float* __restrict__ A2, const float* __restrict__ W2,
    const float* __restrict__ bias, float* __restrict__ C,
    int M, int N, int K, int relu)
{
    __shared__ float As[16][72];    // [k][m]  A tile transposed, padded stride
    __shared__ float Ws[16][136];   // [k][n]  W tile, padded stride

    const int tid  = threadIdx.x;
    const int lane = tid & 31;
    const int wave = tid >> 5;
    const int wm   = wave & 3;      // 16-row slice within 64-row tile
    const int wn   = wave >> 2;     // 64-col slice within 128-col tile
    const int tileM = blockIdx.x * 64;
    const int tileN = blockIdx.y * 128;

    const int ml   = lane & 15;
    const int half = lane >> 4;     // ISA A/B layout: lanes 16-31 hold K+2,K+3

    v8f acc[4];
#pragma unroll
    for (int i = 0; i < 4; ++i)
        acc[i] = (v8f){0.f, 0.f, 0.f, 0.f, 0.f, 0.f, 0.f, 0.f};

    // A staging: thread t loads float4 at (row = t/4, k = (t%4)*4)
    const int ldRow = tid >> 2;        // 0..63
    const int ldK   = (tid & 3) * 4;   // 0,4,8,12
    // W staging: thread t loads 2x float4 at (k = t/32 [+8], n = (t%32)*4)
    const int wK  = tid >> 5;          // 0..7
    const int wN4 = (tid & 31) * 4;    // 0..124

    for (int s = 0; s < 2; ++s) {
        const float* A = s ? A2 : A1;
        const float* W = s ? W2 : W1;
        for (int k0 = 0; k0 < K; k0 += 16) {
            // ---- stage A[tileM..+64][k0..+16] transposed into LDS ----
            {
                int row = tileM + ldRow;
                float4 v = make_float4(0.f, 0.f, 0.f, 0.f);
                if (row < M)
                    v = *(const float4*)(A + (size_t)row * K + k0 + ldK);
                As[ldK + 0][ldRow] = v.x;
                As[ldK + 1][ldRow] = v.y;
                As[ldK + 2][ldRow] = v.z;
                As[ldK + 3][ldRow] = v.w;
            }
            // ---- stage W[k0..+16][tileN..+128] into LDS (coalesced) ----
            {
                float4 w0 = *(const float4*)(W + (size_t)(k0 + wK) * N + tileN + wN4);
                float4 w1 = *(const float4*)(W + (size_t)(k0 + 8 + wK) * N + tileN + wN4);
                *(float4*)&Ws[wK][wN4]     = w0;
                *(float4*)&Ws[wK + 8][wN4] = w1;
            }
            __syncthreads();

#pragma unroll
            for (int kk = 0; kk < 16; kk += 4) {
                const int kr = kk + 2 * half;
                // A frag (16x4): v0 = K+0/K+2, v1 = K+1/K+3 per lane half
                v2f a;
                a.x = As[kr + 0][wm * 16 + ml];
                a.y = As[kr + 1][wm * 16 + ml];
#pragma unroll
                for (int cb = 0; cb < 4; ++cb) {
                    const int col = wn * 64 + cb * 16 + ml;
                    v2f b;
                    b.x = Ws[kr + 0][col];
                    b.y = Ws[kr + 1][col];
                    acc[cb] = __builtin_amdgcn_wmma_f32_16x16x4_f32(
                        false, a, false, b, (short)0, acc[cb], false, false);
                }
            }
            __syncthreads();
        }
    }

    // C/D layout: VGPR i -> row i (lanes 0-15) / row i+8 (lanes 16-31)
#pragma unroll
    for (int cb = 0; cb < 4; ++cb) {
        int col = tileN + wn * 64 + cb * 16 + ml;
        float bc = bias[col];
#pragma unroll
        for (int i = 0; i < 8; ++i) {
            int row = tileM + wm * 16 + i + 8 * half;
            if (row < M) {
                float v = acc[cb][i] + bc;
                if (relu) v = fmaxf(v, 0.f);
                C[(size_t)row * N + col] = v;
            }
        }
    }
}

// ---------------------------------------------------------------------------
// out[l] = -dot(zU[ls[l]], zI[ld[l]]);  out[L+l] = +dot(...)
// One wave per label; float4 per lane (128-d), shuffle reduce (wave32).
// ---------------------------------------------------------------------------
__global__ __launch_bounds__(256) void decode_kernel(
    const float* __restrict__ zU, const float* __restrict__ zI,
    const int* __restrict__ ls, const int* __restrict__ ld,
    float* __restrict__ out, int L)
{
    int wid  = (blockIdx.x * blockDim.x + threadIdx.x) >> 5;
    int lane = threadIdx.x & 31;
    if (wid >= L) return;
    const float4* a = (const float4*)(zU + (size_t)ls[wid] * 128);
    const float4* b = (const float4*)(zI + (size_t)ld[wid] * 128);
    float4 av = a[lane];
    float4 bv = b[lane];
    float s = av.x * bv.x + av.y * bv.y + av.z * bv.z + av.w * bv.w;
#pragma unroll
    for (int off = 16; off > 0; off >>= 1)
        s += __shfl_down(s, off, 32);
    if (lane == 0) {
        out[wid]     = -s;
        out[L + wid] =  s;
    }
}

// ---------------------------------------------------------------------------
extern "C" void kernel_launch(void* const* d_in, const int* in_sizes, int n_in,
                              void* d_out, int out_size, void* d_ws, size_t ws_size,
                              hipStream_t stream)
{
    const float* x_user = (const float*)d_in[0];
    const float* x_item = (const float*)d_in[1];
    const float* Wl1_ui = (const float*)d_in[2];
    const float* Wr1_ui = (const float*)d_in[3];
    const float* b1_ui  = (const float*)d_in[4];
    const float* Wl1_iu = (const float*)d_in[5];
    const float* Wr1_iu = (const float*)d_in[6];
    const float* b1_iu  = (const float*)d_in[7];
    const float* Wl2_ui = (const float*)d_in[8];
    const float* Wr2_ui = (const float*)d_in[9];
    const float* b2_ui  = (const float*)d_in[10];
    const float* Wl2_iu = (const float*)d_in[11];
    const float* Wr2_iu = (const float*)d_in[12];
    const float* b2_iu  = (const float*)d_in[13];
    const int* src_u = (const int*)d_in[14];
    const int* dst_i = (const int*)d_in[15];
    const int* lsrc  = (const int*)d_in[16];
    const int* ldst  = (const int*)d_in[17];

    const int DI = 128, DH = 256;
    const int NU = in_sizes[0] / DI;
    const int NI = in_sizes[1] / DI;
    const int E  = in_sizes[14];
    const int L  = in_sizes[16];

    // workspace layout (floats)
    float* ws = (float*)d_ws;
    size_t o = 0;
    float* aggI  = ws + o; o += (size_t)NI * DI;   // layer-1 agg -> reused as z_item
    float* aggU  = ws + o; o += (size_t)NU * DI;   // layer-1 agg -> reused as z_user
    float* degI  = ws + o; o += (size_t)NI;
    float* degU  = ws + o; o += (size_t)NU;
    float* agg2I = ws + o; o += (size_t)NI * DH;
    float* agg2U = ws + o; o += (size_t)NU * DH;
    size_t zeroFloats = o;                         // everything above is accumulated into
    float* hI    = ws + o; o += (size_t)NI * DH;   // fully overwritten, no zeroing needed
    float* hU    = ws + o; o += (size_t)NU * DH;

    hipMemsetAsync(d_ws, 0, zeroFloats * sizeof(float), stream);

    const int edgeGrid = (E + 7) / 8;   // 8 waves / 256-thread block

    // ---- layer 1: scatter-mean over both edge directions ----
    scatter_edges<<<edgeGrid, 256, 0, stream>>>(
        x_user, x_item, src_u, dst_i, aggI, aggU, degI, degU, E, DI, 1);
    normalize_kernel<<<(NI * (DI / 4) + 255) / 256, 256, 0, stream>>>(aggI, degI, NI, DI);
    normalize_kernel<<<(NU * (DI / 4) + 255) / 256, 256, 0, stream>>>(aggU, degU, NU, DI);

    // ---- layer 1 GEMMs (fused dual mat-mul + bias + relu) ----
    gemm_dual<<<dim3((NI + 63) / 64, DH / 128), 256, 0, stream>>>(
        aggI, Wl1_ui, x_item, Wr1_ui, b1_ui, hI, NI, DH, DI, 1);
    gemm_dual<<<dim3((NU + 63) / 64, DH / 128), 256, 0, stream>>>(
        aggU, Wl1_iu, x_user, Wr1_iu, b1_iu, hU, NU, DH, DI, 1);

    // ---- layer 2: scatter-mean of hidden states (degrees already known) ----
    scatter_edges<<<edgeGrid, 256, 0, stream>>>(
        hU, hI, src_u, dst_i, agg2I, agg2U, degI, degU, E, DH, 0);
    normalize_kernel<<<(NI * (DH / 4) + 255) / 256, 256, 0, stream>>>(agg2I, degI, NI, DH);
    normalize_kernel<<<(NU * (DH / 4) + 255) / 256, 256, 0, stream>>>(agg2U, degU, NU, DH);

    // ---- layer 2 GEMMs (no relu); outputs reuse layer-1 agg buffers ----
    gemm_dual<<<dim3((NI + 63) / 64, DI / 128), 256, 0, stream>>>(
        agg2I, Wl2_ui, hI, Wr2_ui, b2_ui, aggI /*z_item*/, NI, DI, DH, 0);
    gemm_dual<<<dim3((NU + 63) / 64, DI / 128), 256, 0, stream>>>(
        agg2U, Wl2_iu, hU, Wr2_iu, b2_iu, aggU /*z_user*/, NU, DI, DH, 0);

    // ---- decode ----
    decode_kernel<<<(L + 7) / 8, 256, 0, stream>>>(
        aggU, aggI, lsrc, ldst, (float*)d_out, L);
}